// Efficient_Cross_Attention_24326694764640
// MI455X (gfx1250) — compile-verified
//
#include <hip/hip_runtime.h>
#include <hip/hip_bf16.h>
#include <math.h>

// ---------------------------------------------------------------------------
// Efficient cross attention (fp32) for MI455X / gfx1250, wave32 + WMMA f32.
//   n=8, C=256, H=W=64 (hw=4096), HEADS=8, dk=dv=32.
// Matrix math: V_WMMA_F32_16X16X4_F32 (exact fp32 path on CDNA5).
// Each GEMM wave owns a 16x64 output tile (4 accumulators) so one A fragment
// feeds 4 WMMAs per k-step. Context GEMM uses split-K + f32 global atomics.
// Wave-tile indices are scalarized via readfirstlane so tile decomposition
// and loop control stay in SALU (EXEC remains all-1s across every WMMA).
// ---------------------------------------------------------------------------

typedef __attribute__((ext_vector_type(2))) float v2f;
typedef __attribute__((ext_vector_type(8))) float v8f;

#define NB    8
#define C     256
#define HW    4096
#define HEADS 8
#define DHEAD 32

__device__ __forceinline__ v8f wmma_f32_k4(v2f a, v2f b, v8f c) {
    // 8 args: (neg_a, A, neg_b, B, c_mod, C, reuse_a, reuse_b)
    return __builtin_amdgcn_wmma_f32_16x16x4_f32(false, a, false, b,
                                                 (short)0, c, false, false);
}

// threadIdx.x>>5 is wave-uniform, so readfirstlane is safe: forces SGPR.
__device__ __forceinline__ int wave_tile_id() {
    return __builtin_amdgcn_readfirstlane(blockIdx.x * 4 + (threadIdx.x >> 5));
}

// ---------------------------------------------------------------------------
// Y[b][m][:] = W[m][:] . X[b][:][:] + bias[m] (+ res[b][m][:])
// M = K = 256, N = 4096, batches = 8.
// One 16(M) x 64(N) tile per wave, 4 waves/block -> 2048 blocks.
// ---------------------------------------------------------------------------
template <bool HAS_RES>
__global__ __launch_bounds__(128)
void conv1x1_wmma(const float* __restrict__ W, const float* __restrict__ bias,
                  const float* __restrict__ X, float* __restrict__ Y,
                  const float* __restrict__ res) {
    const int lane  = threadIdx.x & 31;
    const int wtile = wave_tile_id();               // 0 .. 8191 (scalar)
    const int nGroups        = HW / 64;             // 64
    const int groupsPerBatch = (C / 16) * nGroups;  // 1024
    const int b  = wtile / groupsPerBatch;
    const int t  = wtile - b * groupsPerBatch;
    const int mt = t / nGroups;
    const int ng = t - mt * nGroups;
    const int m0 = mt * 16, n0 = ng * 64;
    const int row = lane & 15;
    const int hi  = lane >> 4;                      // 0 or 1

    const float* __restrict__ Xb = X + (size_t)b * C * HW + n0 + row;
    const float* __restrict__ Wr = W + (size_t)(m0 + row) * C;

    v8f acc[4] = {};
#pragma unroll 4
    for (int k0 = 0; k0 < C; k0 += 4) {
        const int kk = k0 + 2 * hi;
        v2f a;
        a.x = Wr[kk];
        a.y = Wr[kk + 1];
        const float* x0 = Xb + (size_t)kk * HW;
        const float* x1 = Xb + (size_t)(kk + 1) * HW;
        // speculative prefetch of the panel 16 k-rows ahead (global_prefetch_b8)
        __builtin_prefetch(x0 + (size_t)16 * HW, 0, 1);
#pragma unroll
        for (int j = 0; j < 4; ++j) {
            v2f bb;
            bb.x = x0[j * 16];
            bb.y = x1[j * 16];
            acc[j] = wmma_f32_k4(a, bb, acc[j]);
        }
    }

    float* __restrict__ Yb = Y + (size_t)b * C * HW + (size_t)m0 * HW + n0;
    const float* __restrict__ Rb =
        HAS_RES ? res + (size_t)b * C * HW + (size_t)m0 * HW + n0 : nullptr;
#pragma unroll
    for (int r = 0; r < 8; ++r) {
        const int m = r + 8 * hi;
        const float bi = bias[m0 + m];
#pragma unroll
        for (int j = 0; j < 4; ++j) {
            float v = acc[j][r] + bi;
            if constexpr (HAS_RES) v += Rb[(size_t)m * HW + j * 16 + row];
            Yb[(size_t)m * HW + j * 16 + row] = v;
        }
    }
}

// ---------------------------------------------------------------------------
// Zero a float buffer (ctx accumulators for split-K atomics).
// ---------------------------------------------------------------------------
__global__ __launch_bounds__(256)
void zero_f32(float* __restrict__ p, int n) {
    const int i = blockIdx.x * 256 + threadIdx.x;
    if (i < n) p[i] = 0.0f;
}

// ---------------------------------------------------------------------------
// In-place softmax over the spatial axis (row of length 4096). One block/row.
// ---------------------------------------------------------------------------
__global__ __launch_bounds__(256)
void softmax_rows(float* __restrict__ data) {
    __shared__ float red[256];
    float* row = data + (size_t)blockIdx.x * HW;
    const int tid = threadIdx.x;

    float m = -INFINITY;
    for (int i = tid; i < HW; i += 256) m = fmaxf(m, row[i]);
    red[tid] = m; __syncthreads();
    for (int s = 128; s > 0; s >>= 1) {
        if (tid < s) red[tid] = fmaxf(red[tid], red[tid + s]);
        __syncthreads();
    }
    m = red[0]; __syncthreads();

    float sum = 0.f;
    for (int i = tid; i < HW; i += 256) {
        const float e = __expf(row[i] - m);
        row[i] = e;
        sum += e;
    }
    red[tid] = sum; __syncthreads();
    for (int s = 128; s > 0; s >>= 1) {
        if (tid < s) red[tid] += red[tid + s];
        __syncthreads();
    }
    const float inv = 1.0f / red[0];
    for (int i = tid; i < HW; i += 256) row[i] *= inv;
}

// ---------------------------------------------------------------------------
// In-place softmax over the 32 per-head channels at each spatial position.
// One thread per (batch, head, position): 8*8*4096 = 262144 threads.
// ---------------------------------------------------------------------------
__global__ __launch_bounds__(256)
void softmax_cols32(float* __restrict__ q) {
    const unsigned idx = blockIdx.x * 256u + threadIdx.x;
    const int p = idx & (HW - 1);
    const int h = (idx >> 12) & (HEADS - 1);
    const int b = idx >> 15;
    float* base = q + ((size_t)b * C + (size_t)h * DHEAD) * HW + p;

    float vals[DHEAD];
    float m = -INFINITY;
#pragma unroll
    for (int i = 0; i < DHEAD; ++i) {
        vals[i] = base[(size_t)i * HW];
        m = fmaxf(m, vals[i]);
    }
    float s = 0.f;
#pragma unroll
    for (int i = 0; i < DHEAD; ++i) {
        vals[i] = __expf(vals[i] - m);
        s += vals[i];
    }
    const float inv = 1.0f / s;
#pragma unroll
    for (int i = 0; i < DHEAD; ++i) base[(size_t)i * HW] = vals[i] * inv;
}

// ---------------------------------------------------------------------------
// ctx[b][h][k][v] (32x32) += sum_p key[b][h*32+k][p] * val[b][h*32+v][p]
// M=N=32, K=4096 per (b,h). Split-K 16-way; f32 global atomics combine
// partials. 64 bh * 4 tiles * 16 splits = 4096 waves -> 1024 blocks.
// ctx must be zeroed first.
// ---------------------------------------------------------------------------
#define CTX_SPLITK 16

__global__ __launch_bounds__(128)
void context_wmma(const float* __restrict__ key, const float* __restrict__ val,
                  float* __restrict__ ctx) {
    const int lane  = threadIdx.x & 31;
    const int wtile = wave_tile_id();          // 0..4095 (scalar)
    const int split = wtile & (CTX_SPLITK - 1);
    const int t  = wtile >> 4;                 // 0..255
    const int bh = t >> 2;                     // 0..63
    const int tt = t & 3;
    const int mt = tt >> 1, nt = tt & 1;
    const int b = bh >> 3, h = bh & 7;
    const int row = lane & 15, hi = lane >> 4;

    const float* __restrict__ kb =
        key + ((size_t)b * C + (size_t)h * DHEAD + mt * 16 + row) * HW;
    const float* __restrict__ vb =
        val + ((size_t)b * C + (size_t)h * DHEAD + nt * 16 + row) * HW;

    const int kbeg = split * (HW / CTX_SPLITK);      // 256-wide K slice (scalar)
    const int kend = kbeg + (HW / CTX_SPLITK);

    v8f acc = {};
#pragma unroll 8
    for (int k0 = kbeg; k0 < kend; k0 += 4) {
        const int kk = k0 + 2 * hi;
        v2f a, bb;
        a.x  = kb[kk];
        a.y  = kb[kk + 1];
        bb.x = vb[kk];
        bb.y = vb[kk + 1];
        acc = wmma_f32_k4(a, bb, acc);
    }

    float* cb = ctx + (size_t)bh * (DHEAD * DHEAD) + (mt * 16) * DHEAD + nt * 16;
#pragma unroll
    for (int r = 0; r < 8; ++r)
        unsafeAtomicAdd(&cb[(r + 8 * hi) * DHEAD + row], acc[r]);
}

// ---------------------------------------------------------------------------
// agg[b][h*32+m][p] = sum_k ctx[b][h][k][m] * q[b][h*32+k][p]
// M=32, N=4096, K=32 per (b,h). One 16x64 tile per wave:
// 64 bh * 2 * 64 = 8192 waves -> 2048 blocks.
// ---------------------------------------------------------------------------
__global__ __launch_bounds__(128)
void attended_wmma(const float* __restrict__ ctx, const float* __restrict__ q,
                   float* __restrict__ agg) {
    const int lane  = threadIdx.x & 31;
    const int wtile = wave_tile_id();          // 0..8191 (scalar)
    const int bh = wtile >> 7;                 // 0..63
    const int t  = wtile & 127;
    const int mt = t >> 6;                     // 0..1
    const int ng = t & 63;                     // 0..63
    const int b = bh >> 3, h = bh & 7;
    const int m0 = mt * 16, n0 = ng * 64;
    const int row = lane & 15, hi = lane >> 4;

    const float* __restrict__ cbase = ctx + (size_t)bh * (DHEAD * DHEAD);
    const float* __restrict__ qbase =
        q + ((size_t)b * C + (size_t)h * DHEAD) * HW + n0 + row;

    v8f acc[4] = {};
#pragma unroll
    for (int k0 = 0; k0 < DHEAD; k0 += 4) {
        const int kk = k0 + 2 * hi;
        v2f a;
        a.x = cbase[kk * DHEAD + m0 + row];         // A[m,k] = ctx[k][m]
        a.y = cbase[(kk + 1) * DHEAD + m0 + row];
        const float* q0 = qbase + (size_t)kk * HW;
        const float* q1 = qbase + (size_t)(kk + 1) * HW;
#pragma unroll
        for (int j = 0; j < 4; ++j) {
            v2f bb;
            bb.x = q0[j * 16];
            bb.y = q1[j * 16];
            acc[j] = wmma_f32_k4(a, bb, acc[j]);
        }
    }

    float* ob = agg + ((size_t)b * C + (size_t)h * DHEAD + m0) * HW + n0;
#pragma unroll
    for (int r = 0; r < 8; ++r) {
        const int m = r + 8 * hi;
#pragma unroll
        for (int j = 0; j < 4; ++j)
            ob[(size_t)m * HW + j * 16 + row] = acc[j][r];
    }
}

// ---------------------------------------------------------------------------
// Host-side orchestration
// ---------------------------------------------------------------------------
extern "C" void kernel_launch(void* const* d_in, const int* in_sizes, int n_in,
                              void* d_out, int out_size, void* d_ws, size_t ws_size,
                              hipStream_t stream) {
    const float* in_rgb = (const float*)d_in[0];
    const float* in_ir  = (const float*)d_in[1];
    const float* Wk_rgb = (const float*)d_in[2];
    const float* bk_rgb = (const float*)d_in[3];
    const float* Wk_ir  = (const float*)d_in[4];
    const float* bk_ir  = (const float*)d_in[5];
    const float* Wq_rgb = (const float*)d_in[6];
    const float* bq_rgb = (const float*)d_in[7];
    const float* Wq_ir  = (const float*)d_in[8];
    const float* bq_ir  = (const float*)d_in[9];
    const float* Wv_rgb = (const float*)d_in[10];
    const float* bv_rgb = (const float*)d_in[11];
    const float* Wv_ir  = (const float*)d_in[12];
    const float* bv_ir  = (const float*)d_in[13];
    const float* Wr_rgb = (const float*)d_in[14];
    const float* br_rgb = (const float*)d_in[15];
    const float* Wr_ir  = (const float*)d_in[16];
    const float* br_ir  = (const float*)d_in[17];

    float* ws = (float*)d_ws;
    const size_t SZ = (size_t)NB * C * HW;       // 8,388,608 floats (32 MB)
    float* k_rgb = ws + 0 * SZ;
    float* q_rgb = ws + 1 * SZ;
    float* v_rgb = ws + 2 * SZ;
    float* k_ir  = ws + 3 * SZ;
    float* q_ir  = ws + 4 * SZ;
    float* v_ir  = ws + 5 * SZ;
    float* agg_rgb = ws + 6 * SZ;
    float* agg_ir  = ws + 7 * SZ;
    float* ctxA = ws + 8 * SZ;                   // keys_ir x values_ir
    float* ctxB = ctxA + (size_t)NB * HEADS * DHEAD * DHEAD;
    const int ctxElems = 2 * NB * HEADS * DHEAD * DHEAD;   // 131072 (both ctx)

    float* out_rgb = (float*)d_out;
    float* out_ir  = out_rgb + SZ;

    const int convBlocks = (NB * (C / 16) * (HW / 64)) / 4;            // 2048
    const int attBlocks  = (NB * HEADS * 2 * (HW / 64)) / 4;           // 2048
    const int ctxBlocks  = (NB * HEADS * 4 * CTX_SPLITK) / 4;          // 1024

    // QKV projections (fp32 WMMA GEMMs)
    conv1x1_wmma<false><<<convBlocks, 128, 0, stream>>>(Wk_rgb, bk_rgb, in_rgb, k_rgb, nullptr);
    conv1x1_wmma<false><<<convBlocks, 128, 0, stream>>>(Wq_rgb, bq_rgb, in_rgb, q_rgb, nullptr);
    conv1x1_wmma<false><<<convBlocks, 128, 0, stream>>>(Wv_rgb, bv_rgb, in_rgb, v_rgb, nullptr);
    conv1x1_wmma<false><<<convBlocks, 128, 0, stream>>>(Wk_ir,  bk_ir,  in_ir,  k_ir,  nullptr);
    conv1x1_wmma<false><<<convBlocks, 128, 0, stream>>>(Wq_ir,  bq_ir,  in_ir,  q_ir,  nullptr);
    conv1x1_wmma<false><<<convBlocks, 128, 0, stream>>>(Wv_ir,  bv_ir,  in_ir,  v_ir,  nullptr);

    // Softmaxes (in place)
    softmax_rows<<<NB * C, 256, 0, stream>>>(k_rgb);
    softmax_rows<<<NB * C, 256, 0, stream>>>(k_ir);
    softmax_cols32<<<(NB * HEADS * HW) / 256, 256, 0, stream>>>(q_rgb);
    softmax_cols32<<<(NB * HEADS * HW) / 256, 256, 0, stream>>>(q_ir);

    // context = softmax(K) @ V^T   (split-K, atomically combined)
    zero_f32<<<(ctxElems + 255) / 256, 256, 0, stream>>>(ctxA, ctxElems);
    context_wmma<<<ctxBlocks, 128, 0, stream>>>(k_ir,  v_ir,  ctxA);
    context_wmma<<<ctxBlocks, 128, 0, stream>>>(k_rgb, v_rgb, ctxB);

    // attended = context^T @ softmax(Q)
    attended_wmma<<<attBlocks, 128, 0, stream>>>(ctxA, q_rgb, agg_rgb);
    attended_wmma<<<attBlocks, 128, 0, stream>>>(ctxB, q_ir,  agg_ir);

    // Output projection + residual
    conv1x1_wmma<true><<<convBlocks, 128, 0, stream>>>(Wr_rgb, br_rgb, agg_rgb, out_rgb, in_rgb);
    conv1x1_wmma<true><<<convBlocks, 128, 0, stream>>>(Wr_ir,  br_ir,  agg_ir,  out_ir,  in_ir);
}